// StateSpaceModel_15693810499843
// MI455X (gfx1250) — compile-verified
//
#include <hip/hip_runtime.h>

// MI455X (gfx1250) state-space model: y = (scan of tanh(h A^T + x B^T)) C^T
//
// Phase 1: xB[bt,s]   = sum_h x[bt,h]*Bm[s,h]        (WMMA f32 16x16x4, K=768)
// Phase 2: in-place scan over t (serial), 32 independent batch chains
// Phase 3: y[bt,h]    = sum_s hs[bt,s]*Cm[h,s]       (WMMA f32 16x16x4, K=16)
//
// Memory-bound: ~0.8 GB moved for ~6.5 GFLOP -> HBM roofline ~35us for the
// GEMM phases; the 8 MB xB/hs workspace is L2-resident (192 MB L2).
// The serial scan is the latency bottleneck: per-step critical path is
// ds_bpermute broadcast + short FMA tree + native v_tanh_f32.

typedef __attribute__((ext_vector_type(2))) float v2f;
typedef __attribute__((ext_vector_type(8))) float v8f;

#define BATCH 32
#define TLEN  4096
#define HDIM  768
#define SDIM  16
#define BT    (BATCH * TLEN)   // 131072 rows

static __device__ __forceinline__ float fast_tanh(float x) {
#if __has_builtin(__builtin_amdgcn_tanhf)
    return __builtin_amdgcn_tanhf(x);          // v_tanh_f32 (CDNA5 TRANS op)
#elif __has_builtin(__builtin_amdgcn_tanh_f32)
    return __builtin_amdgcn_tanh_f32(x);
#else
    return tanhf(x);
#endif
}

// ---------------------------------------------------------------------------
// Phase 1: xB = x * B^T.  One wave per 16-row M tile (N = S = 16 = one tile).
// A-frag (16x4 f32): lanes 0-15 hold {K=k,k+1} of row M=lane, lanes 16-31
// hold {K=k+2,k+3}.  B-frag (4x16): lanes 0-15 hold rows K=k,k+1 at N=lane,
// lanes 16-31 rows K=k+2,k+3 at N=lane-16.  (ISA 7.12.2 layouts.)
// ---------------------------------------------------------------------------
__global__ __launch_bounds__(128) void k_xproj(const float* __restrict__ x,
                                               const float* __restrict__ Bm,
                                               float* __restrict__ xB) {
    const int lane = threadIdx.x & 31;
    const int wave = blockIdx.x * (blockDim.x >> 5) + (threadIdx.x >> 5);
    const int half = lane >> 4;      // 0: lanes 0-15, 1: lanes 16-31
    const int l16  = lane & 15;
    const long m0   = (long)wave * 16;
    const long rowA = m0 + l16;      // x row for A fragment
    const long rowB = l16;           // Bm row (== output column n) for B frag

    v8f c = {};
    #pragma unroll 8
    for (int k = 0; k < HDIM; k += 4) {
        const int ka = k + 2 * half;
        v2f a, b;
        a.x = x[rowA * HDIM + ka];
        a.y = x[rowA * HDIM + ka + 1];
        // B^T[k, n] = Bm[n, k]
        b.x = Bm[rowB * HDIM + ka];
        b.y = Bm[rowB * HDIM + ka + 1];
        c = __builtin_amdgcn_wmma_f32_16x16x4_f32(
                /*neg_a=*/false, a, /*neg_b=*/false, b,
                /*c_mod=*/(short)0, c, /*reuse_a=*/false, /*reuse_b=*/false);
    }
    // D layout: VGPR r -> row m0+r (lanes 0-15), row m0+8+r (lanes 16-31)
    const long rbase = m0 + 8 * half;
    #pragma unroll
    for (int r = 0; r < 8; ++r)
        xB[(rbase + r) * SDIM + l16] = c[r];
}

// ---------------------------------------------------------------------------
// Phase 2: h_t = tanh(h_{t-1} A^T + xb_t), overwriting xB with h in place.
// One wave handles TWO batch chains: lane = 16*pair + s.  A's row A[s][*]
// lives in 16 registers per lane; h[k] broadcasts come from wave shuffles
// (ds_bpermute) within each 16-lane half.  Four partial accumulators keep
// the dependent-FMA chain at ~4 deep; the 16 shuffles pipeline underneath.
// 32 chains run concurrently across 16 single-wave blocks.
// ---------------------------------------------------------------------------
__global__ __launch_bounds__(32) void k_scan(const float* __restrict__ Amat,
                                             float* __restrict__ hs) {
    const int lane = threadIdx.x & 31;
    const int s    = lane & 15;
    const int pair = lane >> 4;
    const int b    = blockIdx.x * 2 + pair;

    float arow[16];
    #pragma unroll
    for (int k = 0; k < 16; ++k) arow[k] = Amat[s * 16 + k];

    float* base = hs + (long)b * TLEN * SDIM;
    float h = 0.0f;
    for (int t = 0; t < TLEN; ++t) {
        const float xb = base[t * SDIM + s];
        if (t + 8 < TLEN)  // pull next lines toward the WGP (L2-resident)
            __builtin_prefetch(&base[(t + 8) * SDIM + s], 0, 3);

        // broadcast the 16 h values of this half's batch
        float hk[16];
        #pragma unroll
        for (int k = 0; k < 16; ++k)
            hk[k] = __shfl(h, (pair << 4) | k, 32);

        // 4-way split accumulation: dependent chain = 4 FMAs + 2 adds
        float a0 = xb, a1 = 0.0f, a2 = 0.0f, a3 = 0.0f;
        #pragma unroll
        for (int k = 0; k < 4; ++k) {
            a0 = fmaf(hk[k],      arow[k],      a0);
            a1 = fmaf(hk[k + 4],  arow[k + 4],  a1);
            a2 = fmaf(hk[k + 8],  arow[k + 8],  a2);
            a3 = fmaf(hk[k + 12], arow[k + 12], a3);
        }
        h = fast_tanh((a0 + a1) + (a2 + a3));
        base[t * SDIM + s] = h;
    }
}

// ---------------------------------------------------------------------------
// Phase 3: y = hs * C^T.  One wave per 16x16 output tile; K = S = 16 -> four
// chained wmma_f32_16x16x4.  C (48 KB) is L2/L1 resident and shared by all.
// ---------------------------------------------------------------------------
__global__ __launch_bounds__(128) void k_yproj(const float* __restrict__ hs,
                                               const float* __restrict__ Cm,
                                               float* __restrict__ y) {
    const int lane = threadIdx.x & 31;
    const int wave = blockIdx.x * (blockDim.x >> 5) + (threadIdx.x >> 5);
    const int NT   = HDIM / 16;            // 48 N tiles
    const int  ntile = wave % NT;
    const long mtile = wave / NT;
    const long m0 = mtile * 16;
    const int  n0 = ntile * 16;
    const int half = lane >> 4;
    const int l16  = lane & 15;

    v8f c = {};
    #pragma unroll
    for (int k = 0; k < SDIM; k += 4) {
        const int ka = k + 2 * half;
        v2f a, b;
        const long rowA = m0 + l16;
        a.x = hs[rowA * SDIM + ka];
        a.y = hs[rowA * SDIM + ka + 1];
        // B^T[k, n] = Cm[n, k]
        const long rowC = n0 + l16;
        b.x = Cm[rowC * SDIM + ka];
        b.y = Cm[rowC * SDIM + ka + 1];
        c = __builtin_amdgcn_wmma_f32_16x16x4_f32(
                false, a, false, b, (short)0, c, false, false);
    }
    const long rbase = m0 + 8 * half;
    #pragma unroll
    for (int r = 0; r < 8; ++r)
        y[(rbase + r) * HDIM + n0 + l16] = c[r];
}

// ---------------------------------------------------------------------------
extern "C" void kernel_launch(void* const* d_in, const int* in_sizes, int n_in,
                              void* d_out, int out_size, void* d_ws, size_t ws_size,
                              hipStream_t stream) {
    const float* x  = (const float*)d_in[0];  // [32,4096,768]
    const float* A  = (const float*)d_in[1];  // [16,16]
    const float* Bm = (const float*)d_in[2];  // [16,768]
    const float* Cm = (const float*)d_in[3];  // [768,16]
    float* y  = (float*)d_out;                // [32,4096,768]
    float* xB = (float*)d_ws;                 // [BT,16] = 8 MB, reused as hs

    // Phase 1: 8192 M tiles, 1 wave each -> 2048 blocks x 128 threads
    k_xproj<<<BT / 16 / 4, 128, 0, stream>>>(x, Bm, xB);
    // Phase 2: 16 blocks x 1 wave, 2 batch chains per wave
    k_scan<<<BATCH / 2, 32, 0, stream>>>(A, xB);
    // Phase 3: 8192 x 48 tiles, 1 wave each -> 98304 blocks x 128 threads
    k_yproj<<<(long)(BT / 16) * (HDIM / 16) / 4, 128, 0, stream>>>(xB, Cm, y);
}